// AttnDecoderRNN_27419071218477
// MI455X (gfx1250) — compile-verified
//
#include <hip/hip_runtime.h>
#include <math.h>

// Problem constants
constexpr int H  = 512;
constexpr int V  = 50257;
constexpr int B  = 32;
constexpr int L  = 50;
constexpr int K2H = 2 * H;      // 1024
constexpr int G4H = 4 * H;      // 2048

// Output layout (flat, return order): logp[B*V], h1[B*H], c1[B*H], attn[B*L]
constexpr int OUT_H1   = B * V;
constexpr int OUT_C1   = OUT_H1 + B * H;
constexpr int OUT_ATTN = OUT_C1 + B * H;

typedef float v2f __attribute__((ext_vector_type(2)));
typedef float v4f __attribute__((ext_vector_type(4)));
typedef float v8f __attribute__((ext_vector_type(8)));
typedef int   v4i __attribute__((ext_vector_type(4)));

// CDNA5 async memory->LDS path (GLOBAL_LOAD_ASYNC_TO_LDS_B128, ASYNCcnt)
#if defined(__gfx1250__) && __has_builtin(__builtin_amdgcn_global_load_async_to_lds_b128)
#define USE_ASYNC 1
#else
#define USE_ASYNC 0
#endif

// Builtin signature (from clang diagnostic): param0 is int4 addrspace(1)*,
// param1 is int4 addrspace(3)*, then imm offset, imm cpol.
typedef __attribute__((address_space(1))) v4i as1_v4i;
typedef __attribute__((address_space(3))) v4i as3_v4i;

// Low 32 bits of a generic shared pointer are the LDS byte offset (ISA §10.2:
// LDS aperture check discards upper bits), so integer truncation yields AS3.
__device__ __forceinline__ as3_v4i* lds_cast4(void* p) {
  return (as3_v4i*)(unsigned int)(unsigned long long)p;
}
__device__ __forceinline__ as1_v4i* gbl_cast4(const void* p) {
  return (as1_v4i*)(unsigned long long)p;
}

__device__ __forceinline__ void wait_async0() {
#if USE_ASYNC
#if __has_builtin(__builtin_amdgcn_s_wait_asynccnt)
  __builtin_amdgcn_s_wait_asynccnt(0);
#else
  asm volatile("s_wait_asynccnt 0x0" ::: "memory");
#endif
#endif
}

__device__ __forceinline__ float wave_sum(float v) {
  #pragma unroll
  for (int off = 16; off > 0; off >>= 1) v += __shfl_xor(v, off, 32);
  return v;
}
__device__ __forceinline__ float wave_max(float v) {
  #pragma unroll
  for (int off = 16; off > 0; off >>= 1) v = fmaxf(v, __shfl_xor(v, off, 32));
  return v;
}
__device__ __forceinline__ float sigm(float x) { return 1.0f / (1.0f + __expf(-x)); }

// ---------------------------------------------------------------------------
// Kernel 1: attention energies + softmax + context.
// grid = B blocks, 256 threads (8 waves). Writes attn to d_out and context
// into X[b][512..1023] in scratch.
// ---------------------------------------------------------------------------
__global__ __launch_bounds__(256) void attn_kernel(
    const float* __restrict__ h0, const float* __restrict__ enc,
    float* __restrict__ attn_out, float* __restrict__ X) {
  const int b = blockIdx.x;
  const int tid = threadIdx.x;
  const int wave = tid >> 5, lane = tid & 31;

  __shared__ float sh_h0[H];
  __shared__ float sE[64];
  __shared__ float sAttn[64];

  for (int h = tid; h < H; h += 256) sh_h0[h] = h0[b * H + h];
  __syncthreads();

  // energies: each wave handles l = wave, wave+8, ...
  for (int l = wave; l < L; l += 8) {
    const float* e = enc + ((size_t)l * B + b) * H;
    float p = 0.0f;
    for (int h = lane; h < H; h += 32) p += sh_h0[h] * e[h];
    p = wave_sum(p);
    if (lane == 0) sE[l] = p;
  }
  __syncthreads();

  // softmax over L=50 by wave 0 (lanes cover l and l+32)
  if (wave == 0) {
    float e0 = (lane < L) ? sE[lane] : -INFINITY;
    float e1 = (lane + 32 < L) ? sE[lane + 32] : -INFINITY;
    float m = wave_max(fmaxf(e0, e1));
    float x0 = (lane < L) ? __expf(e0 - m) : 0.0f;
    float x1 = (lane + 32 < L) ? __expf(e1 - m) : 0.0f;
    float s = wave_sum(x0 + x1);
    float inv = 1.0f / s;
    if (lane < L) { sAttn[lane] = x0 * inv; attn_out[b * L + lane] = x0 * inv; }
    if (lane + 32 < L) { sAttn[lane + 32] = x1 * inv; attn_out[b * L + lane + 32] = x1 * inv; }
  }
  __syncthreads();

  // context[h] = sum_l attn[l] * enc[l,b,h]  -> X[b][512+h]
  for (int h = tid; h < H; h += 256) {
    float acc = 0.0f;
    #pragma unroll 5
    for (int l = 0; l < L; ++l) acc += sAttn[l] * enc[((size_t)l * B + b) * H + h];
    X[b * K2H + H + h] = acc;
  }
}

// ---------------------------------------------------------------------------
// Kernel 2: LSTM gates = rnn_in @ W_ih^T + h0 @ W_hh^T + biases.
// grid = 256 blocks x 256 threads. wave -> one gate row j, lanes -> batch.
// Weight addresses are wave-uniform (broadcast loads); read exactly once.
// ---------------------------------------------------------------------------
__global__ __launch_bounds__(256) void gates_kernel(
    const int* __restrict__ input_data, const float* __restrict__ emb,
    const float* __restrict__ X, const float* __restrict__ h0,
    const float* __restrict__ W_ih, const float* __restrict__ W_hh,
    const float* __restrict__ b_ih, const float* __restrict__ b_hh,
    float* __restrict__ gates) {
  const int tid = threadIdx.x;
  const int b = tid & 31;
  const int j = blockIdx.x * 8 + (tid >> 5);

  const int tok = input_data[b];
  const v4f* e  = (const v4f*)(emb + (size_t)tok * H);      // embedded (first 512 of rnn_in)
  const v4f* cx = (const v4f*)(X + b * K2H + H);            // context  (second 512)
  const v4f* hh = (const v4f*)(h0 + b * H);
  const v4f* wi = (const v4f*)(W_ih + (size_t)j * K2H);
  const v4f* wh = (const v4f*)(W_hh + (size_t)j * H);

  float acc = b_ih[j] + b_hh[j];
  #pragma unroll 4
  for (int k = 0; k < H / 4; ++k) {
    v4f a = e[k], w = wi[k];
    acc += a[0] * w[0] + a[1] * w[1] + a[2] * w[2] + a[3] * w[3];
  }
  #pragma unroll 4
  for (int k = 0; k < H / 4; ++k) {
    v4f a = cx[k], w = wi[H / 4 + k];
    acc += a[0] * w[0] + a[1] * w[1] + a[2] * w[2] + a[3] * w[3];
  }
  #pragma unroll 4
  for (int k = 0; k < H / 4; ++k) {
    v4f a = hh[k], w = wh[k];
    acc += a[0] * w[0] + a[1] * w[1] + a[2] * w[2] + a[3] * w[3];
  }
  gates[b * G4H + j] = acc;
}

// ---------------------------------------------------------------------------
// Kernel 3: combine gates -> c1, h1; write outputs and X[b][0..511] = h1.
// ---------------------------------------------------------------------------
__global__ __launch_bounds__(256) void lstm_combine_kernel(
    const float* __restrict__ gates, const float* __restrict__ c0,
    float* __restrict__ h1_out, float* __restrict__ c1_out,
    float* __restrict__ X) {
  const int idx = blockIdx.x * 256 + threadIdx.x;   // B*H threads
  const int b = idx >> 9, h = idx & 511;
  const float ig = gates[b * G4H + h];
  const float fg = gates[b * G4H + H + h];
  const float gg = gates[b * G4H + 2 * H + h];
  const float og = gates[b * G4H + 3 * H + h];
  const float c1 = sigm(fg) * c0[b * H + h] + sigm(ig) * tanhf(gg);
  const float h1 = sigm(og) * tanhf(c1);
  c1_out[b * H + h] = c1;
  h1_out[b * H + h] = h1;
  X[b * K2H + h] = h1;
}

// ---------------------------------------------------------------------------
// Kernel 4: output projection via V_WMMA_F32_16X16X4_F32 with async
// GLOBAL_LOAD_ASYNC_TO_LDS_B128 double-buffered staging.
// logits[32, V] = X[32,1024] @ W_out^T + b_out.
// Block: 256 threads (8 waves), covers 128 vocab columns; wave owns a 16-wide
// N tile with two accumulators (batch rows 0-15 and 16-31).
// WSTRIDE=36 floats: 16B-aligned rows (async B128), conflict-free b64 reads.
// ---------------------------------------------------------------------------
#define WSTRIDE 36
#define KCHUNK 32
#define NCHUNK (K2H / KCHUNK)

__device__ __forceinline__ void stage_tile(
    const float* __restrict__ Wout, const float* __restrict__ X,
    float* sWbuf, float* sXbuf, int v0, int k0, int tid) {
#if USE_ASYNC
  #pragma unroll
  for (int i = 0; i < 4; ++i) {
    const int r = (tid >> 3) + i * 32;
    const int c = (tid & 7) * 4;
    const int v = v0 + r;
    if (v < V) {
      const float* g = Wout + (size_t)v * K2H + k0 + c;
      __builtin_amdgcn_global_load_async_to_lds_b128(
          gbl_cast4(g), lds_cast4(&sWbuf[r * WSTRIDE + c]), 0, 0);
      if (k0 + KCHUNK < K2H) __builtin_prefetch(g + KCHUNK, 0, 1);
    }
  }
  {
    const int r = tid >> 3;
    const int c = (tid & 7) * 4;
    __builtin_amdgcn_global_load_async_to_lds_b128(
        gbl_cast4(X + r * K2H + k0 + c), lds_cast4(&sXbuf[r * WSTRIDE + c]), 0, 0);
  }
#else
  #pragma unroll
  for (int i = 0; i < 4; ++i) {
    const int r = (tid >> 3) + i * 32;
    const int c = (tid & 7) * 4;
    const int v = v0 + r;
    v4f w = {};
    if (v < V) {
      w = __builtin_nontemporal_load((const v4f*)(Wout + (size_t)v * K2H + k0 + c));
      if (k0 + KCHUNK < K2H)
        __builtin_prefetch(Wout + (size_t)v * K2H + k0 + KCHUNK + c, 0, 1);
    }
    v2f* dst = (v2f*)(&sWbuf[r * WSTRIDE + c]);
    dst[0] = (v2f){w[0], w[1]};
    dst[1] = (v2f){w[2], w[3]};
  }
  {
    const int r = tid >> 3;
    const int c = (tid & 7) * 4;
    v4f x = *(const v4f*)(X + r * K2H + k0 + c);
    v2f* dst = (v2f*)(&sXbuf[r * WSTRIDE + c]);
    dst[0] = (v2f){x[0], x[1]};
    dst[1] = (v2f){x[2], x[3]};
  }
#endif
}

__global__ __launch_bounds__(256) void proj_kernel(
    const float* __restrict__ X, const float* __restrict__ Wout,
    const float* __restrict__ bout, float* __restrict__ logits) {
  __shared__ float sW[2][128 * WSTRIDE];
  __shared__ float sX[2][32 * WSTRIDE];

  const int tid = threadIdx.x;
  const int wave = tid >> 5, lane = tid & 31;
  const int v0 = blockIdx.x * 128;

  v8f acc0 = {};
  v8f acc1 = {};

  // Tail block: zero rows whose vocab index is out of range (async staging
  // predicates those lanes off, so the LDS would otherwise be uninitialized).
  if (v0 + 128 > V) {
    if (tid < 128 && v0 + tid >= V) {
      #pragma unroll
      for (int c = 0; c < KCHUNK; ++c) {
        sW[0][tid * WSTRIDE + c] = 0.0f;
        sW[1][tid * WSTRIDE + c] = 0.0f;
      }
    }
  }

  stage_tile(Wout, X, sW[0], sX[0], v0, 0, tid);

  const int nl = lane & 15;
  const int khalf = (lane >> 4) * 2;   // K offset 0 or 2 per lane half

  for (int kc = 0; kc < NCHUNK; ++kc) {
    wait_async0();        // drain this wave's async copies (prev stage)
    __syncthreads();      // all waves' copies now visible
    if (kc + 1 < NCHUNK)  // overlap next chunk's copies with this chunk's WMMAs
      stage_tile(Wout, X, sW[(kc + 1) & 1], sX[(kc + 1) & 1], v0,
                 (kc + 1) * KCHUNK, tid);

    const float* w = sW[kc & 1];
    const float* x = sX[kc & 1];
    #pragma unroll
    for (int s = 0; s < KCHUNK / 4; ++s) {
      const int k = s * 4 + khalf;
      // B fragment: lane -> N = nl (vocab col within wave tile), K = k, k+1
      v2f bf = *(const v2f*)(w + (wave * 16 + nl) * WSTRIDE + k);
      // A fragments: lane -> M = nl (batch row), K = k, k+1
      v2f a0 = *(const v2f*)(x + nl * WSTRIDE + k);
      v2f a1 = *(const v2f*)(x + (16 + nl) * WSTRIDE + k);
      acc0 = __builtin_amdgcn_wmma_f32_16x16x4_f32(
          false, a0, false, bf, (short)0, acc0, false, false);
      acc1 = __builtin_amdgcn_wmma_f32_16x16x4_f32(
          false, a1, false, bf, (short)0, acc1, false, false);
    }
  }

  // ---- epilogue: C/D layout: VGPR r, lane<16 -> M=r, lane>=16 -> M=8+r; N=lane&15
  const int n = v0 + wave * 16 + nl;
  if (n < V) {
    const float bo = bout[n];
    const int mh = 8 * (lane >> 4);
    #pragma unroll
    for (int r = 0; r < 8; ++r) {
      const int m = r + mh;
      logits[(size_t)m * V + n] = acc0[r] + bo;
      logits[(size_t)(m + 16) * V + n] = acc1[r] + bo;
    }
  }
}

// ---------------------------------------------------------------------------
// Kernel 5: in-place log_softmax per batch row over V.
// ---------------------------------------------------------------------------
__global__ __launch_bounds__(256) void logsoftmax_kernel(float* __restrict__ logp) {
  const int b = blockIdx.x;
  float* row = logp + (size_t)b * V;
  const int tid = threadIdx.x;
  __shared__ float red[256];

  float m = -INFINITY;
  for (int v = tid; v < V; v += 256) m = fmaxf(m, row[v]);
  red[tid] = m;
  __syncthreads();
  for (int s = 128; s > 0; s >>= 1) {
    if (tid < s) red[tid] = fmaxf(red[tid], red[tid + s]);
    __syncthreads();
  }
  m = red[0];
  __syncthreads();

  float sum = 0.0f;
  for (int v = tid; v < V; v += 256) sum += __expf(row[v] - m);
  red[tid] = sum;
  __syncthreads();
  for (int s = 128; s > 0; s >>= 1) {
    if (tid < s) red[tid] += red[tid + s];
    __syncthreads();
  }
  const float lse = m + __logf(red[0]);

  for (int v = tid; v < V; v += 256) row[v] -= lse;
}

// ---------------------------------------------------------------------------
extern "C" void kernel_launch(void* const* d_in, const int* in_sizes, int n_in,
                              void* d_out, int out_size, void* d_ws, size_t ws_size,
                              hipStream_t stream) {
  const int*   input_data = (const int*)d_in[0];
  const float* h0    = (const float*)d_in[1];
  const float* c0    = (const float*)d_in[2];
  const float* enc   = (const float*)d_in[3];
  const float* emb   = (const float*)d_in[4];
  const float* W_ih  = (const float*)d_in[5];
  const float* W_hh  = (const float*)d_in[6];
  const float* b_ih  = (const float*)d_in[7];
  const float* b_hh  = (const float*)d_in[8];
  const float* W_out = (const float*)d_in[9];
  const float* b_out = (const float*)d_in[10];

  float* out = (float*)d_out;
  float* X     = (float*)d_ws;        // [B, 2H] = [h1 | context], 32768 floats
  float* gates = X + B * K2H;         // [B, 4H], 65536 floats

  attn_kernel<<<B, 256, 0, stream>>>(h0, enc, out + OUT_ATTN, X);
  gates_kernel<<<G4H / 8, 256, 0, stream>>>(input_data, emb, X, h0,
                                            W_ih, W_hh, b_ih, b_hh, gates);
  lstm_combine_kernel<<<(B * H) / 256, 256, 0, stream>>>(gates, c0,
                                                         out + OUT_H1, out + OUT_C1, X);
  proj_kernel<<<(V + 127) / 128, 256, 0, stream>>>(X, W_out, b_out, out);
  logsoftmax_kernel<<<B, 256, 0, stream>>>(out);
}